// kMeanClusterStep_30829275251215
// MI455X (gfx1250) — compile-verified
//
#include <hip/hip_runtime.h>
#include <hip/hip_bf16.h>

// k-means assignment + segment-sum step for MI455X (gfx1250, wave32).
//
// score_k(x) = c_k^2/2 - x.c_k   (argmin of this == argmin of d2)
// Cross term computed with V_WMMA_F32_16X16X4_F32 (fp32, matches reference
// precision). x (410MB) streams from HBM once for the GEMM and is re-read
// from L2/WGP$ for accumulation; centroids live in LDS as pre-swizzled
// WMMA B-fragments.

typedef __attribute__((ext_vector_type(2))) float v2f;
typedef __attribute__((ext_vector_type(8))) float v8f;

#define D_DIM   256
#define KPAD    64          // K padded to 4 groups of 16
#define NGRP    4
#define KSTEPS  64          // D / 4 (k-dim per WMMA = 4)
#define BLOCK   256
#define WPB     8           // waves per block (wave32)
#define CHUNK   32          // points per wave per iteration (2 WMMA tiles)

__global__ __launch_bounds__(BLOCK)
void kmeans_step_kernel(const float* __restrict__ x,
                        const float* __restrict__ c,
                        float* __restrict__ outSums,
                        int* __restrict__ outCounts,
                        int N, int K)
{
    // ---- LDS: 64KB B-frags + 64KB sums + misc ~= 132KB ----
    __shared__ v2f      lsB[NGRP * KSTEPS * 32];   // pre-swizzled B fragments
    __shared__ float    lsSums[KPAD * D_DIM];      // per-block partial sums
    __shared__ float    lsC2h[KPAD];               // c_k^2 / 2 (+INF for pad)
    __shared__ unsigned lsCounts[KPAD];
    __shared__ int      lsIdx[WPB * CHUNK];        // per-wave assignment slots

    const int tid = threadIdx.x;

    // ---- block setup ----
    for (int i = tid; i < KPAD * D_DIM; i += BLOCK) lsSums[i] = 0.0f;
    if (tid < KPAD) lsCounts[tid] = 0u;

    // B fragment for (group g, k-step s, lane l):
    //   n = (l&15) + 16g,  k = 4s + 2*(l>>4)   -> float2 {c[n][k], c[n][k+1]}
    // stored linearly by lane => ds_load_b64 hits all 64 banks exactly once.
    for (int i = tid; i < NGRP * KSTEPS * 32; i += BLOCK) {
        int l = i & 31, s = (i >> 5) & 63, g = i >> 11;
        int n = (l & 15) + 16 * g;
        int k = 4 * s + 2 * (l >> 4);
        v2f b; b.x = 0.0f; b.y = 0.0f;
        if (n < K) { b.x = c[n * D_DIM + k]; b.y = c[n * D_DIM + k + 1]; }
        lsB[i] = b;
    }

    if (tid < KPAD) {
        if (tid < K) {
            float acc = 0.0f;
            for (int d = 0; d < D_DIM; ++d) {
                float t = c[tid * D_DIM + d];
                acc += t * t;
            }
            lsC2h[tid] = 0.5f * acc;
        } else {
            lsC2h[tid] = __builtin_inff();   // padded clusters never win
        }
    }
    __syncthreads();

    const int lane     = tid & 31;
    const int l16      = lane & 15;
    const int h        = lane >> 4;
    const int waveSlot = tid >> 5;
    const int waveId   = (blockIdx.x * BLOCK + tid) >> 5;
    const int numWaves = (gridDim.x * BLOCK) >> 5;

    float c2h[NGRP];
#pragma unroll
    for (int g = 0; g < NGRP; ++g) c2h[g] = lsC2h[l16 + 16 * g];

    const int nChunks = N / CHUNK;   // N = 400000 -> 12500 exactly
    for (int chunk = waveId; chunk < nChunks; chunk += numWaves) {
        const int p0 = chunk * CHUNK;
        // A-fragment base for the two 16-point tiles; lane holds row l16,
        // k-pair (k0+2h, k0+2h+1) per ISA 32-bit A layout.
        const float* xa0 = x + (size_t)(p0 + l16) * D_DIM + 2 * h;
        const float* xa1 = xa0 + 16 * D_DIM;

        v8f acc[2][NGRP];
#pragma unroll
        for (int T = 0; T < 2; ++T)
#pragma unroll
            for (int g = 0; g < NGRP; ++g) acc[T][g] = (v8f)0.0f;

        const v2f* bbase = lsB + lane;
#pragma unroll 4
        for (int s = 0; s < KSTEPS; ++s) {
            v2f a0 = *(const v2f*)(xa0 + 4 * s);
            v2f a1 = *(const v2f*)(xa1 + 4 * s);
#pragma unroll
            for (int g = 0; g < NGRP; ++g) {
                v2f b = bbase[(g * KSTEPS + s) * 32];
                acc[0][g] = __builtin_amdgcn_wmma_f32_16x16x4_f32(
                    false, a0, false, b, (short)0, acc[0][g], false, false);
                acc[1][g] = __builtin_amdgcn_wmma_f32_16x16x4_f32(
                    false, a1, false, b, (short)0, acc[1][g], false, false);
            }
        }

        // ---- per-point argmin ----
        // C/D layout: lane = column N (cluster l16+16g), VGPR v = row v + 8h.
#pragma unroll
        for (int T = 0; T < 2; ++T) {
#pragma unroll
            for (int v = 0; v < 8; ++v) {
                float bv = c2h[0] - acc[T][0][v];
                int   bi = l16;
#pragma unroll
                for (int g = 1; g < NGRP; ++g) {
                    float sc = c2h[g] - acc[T][g][v];
                    int   ci = l16 + 16 * g;
                    if (sc < bv || (sc == bv && ci < bi)) { bv = sc; bi = ci; }
                }
#pragma unroll
                for (int off = 1; off < 16; off <<= 1) {
                    float ov = __shfl_xor(bv, off, 16);
                    int   oi = __shfl_xor(bi, off, 16);
                    if (ov < bv || (ov == bv && oi < bi)) { bv = ov; bi = oi; }
                }
                if (l16 == 0)
                    lsIdx[waveSlot * CHUNK + 16 * T + v + 8 * h] = bi;
            }
        }

        // ---- accumulate (LDS atomics; x re-read hits WGP$/L2) ----
        int myIdx = lsIdx[waveSlot * CHUNK + lane];
        atomicAdd(&lsCounts[myIdx], 1u);

#pragma unroll 4
        for (int p = 0; p < CHUNK; ++p) {
            int idx = __shfl(myIdx, p, 32);
            const float* xp = x + (size_t)(p0 + p) * D_DIM;
            float* srow = lsSums + idx * D_DIM;
#pragma unroll
            for (int j = 0; j < 8; ++j) {
                int col = lane + 32 * j;   // conflict-free bank mapping
                atomicAdd(&srow[col], xp[col]);
            }
        }
    }

    __syncthreads();

    // ---- flush block partials to global (d_out pre-zeroed by memset) ----
    for (int i = tid; i < KPAD * D_DIM; i += BLOCK) {
        int n = i >> 8;
        if (n < K) {
            float val = lsSums[i];
            if (val != 0.0f) atomicAdd(&outSums[n * D_DIM + (i & 255)], val);
        }
    }
    if (tid < K) {
        unsigned cnt = lsCounts[tid];
        if (cnt) atomicAdd(&outCounts[tid], (int)cnt);
    }
}

extern "C" void kernel_launch(void* const* d_in, const int* in_sizes, int n_in,
                              void* d_out, int out_size, void* d_ws, size_t ws_size,
                              hipStream_t stream) {
    (void)n_in; (void)d_ws; (void)ws_size;

    const float* x = (const float*)d_in[0];   // locF: (N,1,256) fp32
    const float* c = (const float*)d_in[1];   // Ck:   (1,50,256) fp32

    const int N = in_sizes[0] / D_DIM;        // 400000
    const int K = in_sizes[1] / D_DIM;        // 50

    // Output: sums (K*D floats) then counts (K int32 bit-patterns).
    float* outSums   = (float*)d_out;
    int*   outCounts = (int*)((float*)d_out + (size_t)K * D_DIM);

    // Zero accumulation targets (capture-legal memset node).
    hipMemsetAsync(d_out, 0, (size_t)out_size * sizeof(float), stream);

    const int blocks = 256;   // grid-stride; ~132KB LDS -> up to 2 blocks/WGP
    hipLaunchKernelGGL(kmeans_step_kernel, dim3(blocks), dim3(BLOCK), 0, stream,
                       x, c, outSums, outCounts, N, K);
}